// MultiScaleAttentionBlock_37529424233095
// MI455X (gfx1250) — compile-verified
//
#include <hip/hip_runtime.h>
#include <hip/hip_bf16.h>
#include <cstdint>

// ---------------------------------------------------------------------------
// MultiScaleAttentionBlock for MI455X (gfx1250), B=8 C=256 H=W=224, fp32.
// Bandwidth-bound: min traffic = 2 reads of x + 1 write ~= 1.27 GB @ 23.3 TB/s.
//   K1: one NT read of x -> avg partial sums (per b,c) AND spatial sigmoid map
//   K2: squeeze-excite MLP on WMMA f32 16x16x4 (v_wmma path)
//   K3: one NT read of x (+6% halo) -> sobel edge + fused multiply -> NT write
// x (411 MB) cannot be cached across kernels in 192 MB L2 and is single-touch
// within each kernel -> stream it NT; keep L2 for spatial (1.6 MB, K1->K3
// reuse) and weights. No atomics anywhere -> deterministic.
// ---------------------------------------------------------------------------

typedef __attribute__((ext_vector_type(2))) float v2f;
typedef __attribute__((ext_vector_type(4))) float v4f;
typedef __attribute__((ext_vector_type(8))) float v8f;

#define B_   8
#define C_   256
#define H_   224
#define W_   224
#define HW   (H_ * W_)          // 50176
#define PIX_PER_BLK 1024
#define BLKS_PER_B  (HW / PIX_PER_BLK)   // 49
#define NPART       (BLKS_PER_B * 8)     // 392 partials per (b,c)

__device__ __forceinline__ float sigmoidf(float v) {
    return 1.0f / (1.0f + __expf(-v));
}

// ---------------------------------------------------------------------------
// K1: for each (b, 1024-pixel chunk), stream all 256 channels once (NT).
//   - per-thread v4f accumulator for the spatial dot  sum_c x*w[c]
//   - per-channel wave32 shfl_xor reduction -> partial sums for avg
// ---------------------------------------------------------------------------
__global__ __launch_bounds__(256)
void k_reduce_spatial(const float* __restrict__ x,
                      const float* __restrict__ sw,     // (C)
                      const float* __restrict__ sb,     // (1)
                      float* __restrict__ spatial,      // (B,HW) sigmoided
                      float* __restrict__ partials)     // [B][49][8][C]
{
    const int tid  = threadIdx.x;
    const int blk  = blockIdx.x;            // 0..48
    const int b    = blockIdx.y;            // 0..7
    const int lane = tid & 31;
    const int wv   = tid >> 5;              // 0..7 (wave32)
    const int base = blk * PIX_PER_BLK + tid * 4;

    const float* xb = x + (size_t)b * C_ * HW + base;
    float* pout = partials + (((size_t)(b * BLKS_PER_B + blk)) * 8 + wv) * C_;

    v4f acc = {0.f, 0.f, 0.f, 0.f};

    for (int c = 0; c < C_; ++c) {
        // non-temporal: x is single-touch in this kernel, don't pollute L2
        const v4f v = __builtin_nontemporal_load(
            (const v4f*)(xb + (size_t)c * HW));
        // gfx1250 global_prefetch_b8 for the channel plane 6 ahead (200KB stride)
        const int cp = (c + 6 < C_) ? (c + 6) : c;
        __builtin_prefetch(xb + (size_t)cp * HW, 0, 3);

        const float wgt = sw[c];            // uniform -> scalar load
        acc += v * wgt;

        // deterministic wave32 reduction of this channel's 128 pixels
        float s = (v.x + v.y) + (v.z + v.w);
        #pragma unroll
        for (int off = 16; off >= 1; off >>= 1)
            s += __shfl_xor(s, off, 32);
        if (lane == 0) pout[c] = s;
    }

    const float bias = sb[0];
    v4f o;
    o.x = sigmoidf(acc.x + bias);
    o.y = sigmoidf(acc.y + bias);
    o.z = sigmoidf(acc.z + bias);
    o.w = sigmoidf(acc.w + bias);
    // spatial IS reused (K3 reads it) -> regular temporal store
    *(v4f*)(spatial + (size_t)b * HW + base) = o;
}

// ---------------------------------------------------------------------------
// K2: channel MLP on WMMA f32 16x16x4.
//   h = relu(avg @ cc1_w.T + b1)        (8x256)@(256x64)  -> waves 0..3
//   channel = sigmoid(h @ cc2_w.T + b2) (8x64)@(64x256)   -> waves 0..15
// A operand: M = lane&15; VGPR pair holds K= {0,1} (lanes 0-15) / {2,3}.
// B operand: N = lane&15; same K split across lane halves.
// D: VGPR r -> M = r + (lane>=16 ? 8 : 0), N = lane&15.
// ---------------------------------------------------------------------------
__global__ __launch_bounds__(512)
void k_mlp(const float* __restrict__ partials,   // [B][49][8][C]
           const float* __restrict__ w1,         // (64,256) row-major
           const float* __restrict__ b1,         // (64)
           const float* __restrict__ w2,         // (256,64) row-major
           const float* __restrict__ b2,         // (256)
           float* __restrict__ channel)          // (B,C) sigmoided
{
    __shared__ float avg_s[16][C_];   // M padded 8 -> 16 with zeros
    __shared__ float h_s[16][64];

    const int tid = threadIdx.x;

    // stage 0: finalize avg = sum(partials)/HW (deterministic serial sums)
    for (int e = tid; e < 16 * C_; e += 512) {
        const int m = e >> 8, c = e & (C_ - 1);
        float s = 0.f;
        if (m < B_) {
            const float* p = partials + (size_t)m * NPART * C_ + c;
            for (int k = 0; k < NPART; ++k)
                s += p[(size_t)k * C_];
            s *= (1.0f / (float)HW);
        }
        avg_s[m][c] = s;
    }
    __syncthreads();

    const int lane = tid & 31;
    const int wv   = tid >> 5;
    const int m    = lane & 15;
    const int koff = (lane >> 4) << 1;     // 0 or 2

    // stage 1: h (16x64), 4 waves, K=256
    if (tid < 128) {
        const int n = wv * 16 + (lane & 15);
        v8f acc = {0.f, 0.f, 0.f, 0.f, 0.f, 0.f, 0.f, 0.f};
        for (int kb = 0; kb < 256; kb += 4) {
            v2f a, bm;
            a.x  = avg_s[m][kb + koff];
            a.y  = avg_s[m][kb + koff + 1];
            bm.x = w1[n * 256 + kb + koff];
            bm.y = w1[n * 256 + kb + koff + 1];
            acc = __builtin_amdgcn_wmma_f32_16x16x4_f32(
                false, a, false, bm, (short)0, acc, false, false);
        }
        const float bias = b1[n];
        const int mbase = (lane >> 4) << 3;    // 0 or 8
        #pragma unroll
        for (int r = 0; r < 8; ++r) {
            const float hv = acc[r] + bias;
            h_s[r + mbase][n] = hv > 0.f ? hv : 0.f;
        }
    }
    __syncthreads();

    // stage 2: channel (16x256), 16 waves, K=64
    {
        const int n = wv * 16 + (lane & 15);
        v8f acc = {0.f, 0.f, 0.f, 0.f, 0.f, 0.f, 0.f, 0.f};
        for (int kb = 0; kb < 64; kb += 4) {
            v2f a, bm;
            a.x  = h_s[m][kb + koff];
            a.y  = h_s[m][kb + koff + 1];
            bm.x = w2[n * 64 + kb + koff];
            bm.y = w2[n * 64 + kb + koff + 1];
            acc = __builtin_amdgcn_wmma_f32_16x16x4_f32(
                false, a, false, bm, (short)0, acc, false, false);
        }
        if (lane < 16) {                       // M = r in 0..7 = batches
            const float bias = b2[n];
            #pragma unroll
            for (int r = 0; r < 8; ++r)
                channel[(size_t)r * C_ + n] = sigmoidf(acc[r] + bias);
        }
    }
}

// ---------------------------------------------------------------------------
// K3: per (b,c,32-row group): LDS tile with 1-row/1-col zero halo, sobel
// edge = 9*center - 3x3 boxsum, fused multiply, single NT write of out.
// ---------------------------------------------------------------------------
#define RG   32                 // output rows per block (224/32 = 7 groups)
#define LDSW 232                // 4 + 224 + halo, 16B-aligned rows

__global__ __launch_bounds__(256)
void k_final(const float* __restrict__ x,
             const float* __restrict__ spatial,   // (B,HW) sigmoided
             const float* __restrict__ channel,   // (B,C) sigmoided
             float* __restrict__ out)
{
    __shared__ float t[RG + 2][LDSW];
    const int tid = threadIdx.x;
    const int rg  = blockIdx.x;          // 0..6
    const int c   = blockIdx.y;          // 0..255
    const int b   = blockIdx.z;          // 0..7
    const int r0  = rg * RG;

    const float* plane = x + ((size_t)(b * C_ + c)) * HW;

    // zero halo columns (cols 3 and 228)
    for (int i = tid; i < RG + 2; i += 256) {
        t[i][3]      = 0.f;
        t[i][4 + W_] = 0.f;
    }
    // fill interior (zero rows outside the image => conv zero padding);
    // NT loads: each x element is touched once, halo reuse lives in LDS
    for (int i = tid; i < (RG + 2) * (W_ / 4); i += 256) {
        const int ri = i / (W_ / 4);
        const int ch = i - ri * (W_ / 4);
        const int h  = r0 - 1 + ri;
        v4f v = {0.f, 0.f, 0.f, 0.f};
        if (h >= 0 && h < H_)
            v = __builtin_nontemporal_load(
                (const v4f*)(plane + (size_t)h * W_ + ch * 4));
        *(v4f*)&t[ri][4 + ch * 4] = v;
    }
    __syncthreads();

    const float ch_g = channel[b * C_ + c];                  // uniform
    const float* sp_b = spatial + (size_t)b * HW + (size_t)r0 * W_;
    float* ob = out + ((size_t)(b * C_ + c)) * HW + (size_t)r0 * W_;

    for (int k = 0; k < (RG * W_) / 256; ++k) {
        const int idx = tid + k * 256;
        const int r   = idx / W_;
        const int wc  = idx - r * W_;
        const float c00 = t[r][3 + wc],   c01 = t[r][4 + wc],   c02 = t[r][5 + wc];
        const float c10 = t[r+1][3 + wc], ctr = t[r+1][4 + wc], c12 = t[r+1][5 + wc];
        const float c20 = t[r+2][3 + wc], c21 = t[r+2][4 + wc], c22 = t[r+2][5 + wc];
        const float box  = ((c00 + c01) + (c02 + c10)) + ((ctr + c12) + (c20 + c21)) + c22;
        const float edge = 9.0f * ctr - box;
        const float o = ctr * sp_b[idx] * ch_g * sigmoidf(edge);
        // output is write-once: non-temporal store, don't occupy L2
        __builtin_nontemporal_store(o, ob + idx);
    }
}

// ---------------------------------------------------------------------------
extern "C" void kernel_launch(void* const* d_in, const int* in_sizes, int n_in,
                              void* d_out, int out_size, void* d_ws, size_t ws_size,
                              hipStream_t stream) {
    const float* x  = (const float*)d_in[0];   // (8,256,224,224)
    const float* sw = (const float*)d_in[1];   // (1,256)
    const float* sb = (const float*)d_in[2];   // (1,)
    const float* w1 = (const float*)d_in[3];   // (64,256)
    const float* b1 = (const float*)d_in[4];   // (64,)
    const float* w2 = (const float*)d_in[5];   // (256,64)
    const float* b2 = (const float*)d_in[6];   // (256,)
    float* out = (float*)d_out;

    // workspace layout (all fully written before being read; no pre-zero)
    float* spatial  = (float*)d_ws;                               // B*HW
    float* partials = spatial  + (size_t)B_ * HW;                 // B*49*8*C
    float* channel  = partials + (size_t)B_ * NPART * C_;         // B*C

    k_reduce_spatial<<<dim3(BLKS_PER_B, B_), 256, 0, stream>>>(
        x, sw, sb, spatial, partials);
    k_mlp<<<1, 512, 0, stream>>>(partials, w1, b1, w2, b2, channel);
    k_final<<<dim3(H_ / RG, C_, B_), 256, 0, stream>>>(
        x, spatial, channel, out);
}